// SSKernelDiag_37993280700535
// MI455X (gfx1250) — compile-verified
//
#include <hip/hip_runtime.h>
#include <math.h>

// S4D diagonal SSM kernel generation for MI455X (gfx1250, wave32).
//
// K[h, 128*l1 + l0] = Re( A[l1,:] @ W[:,l0] ),  A = 2*Ctil_n*w_n^(128*l1),
// W = w_n^(l0).  Two f32 WMMA matmuls (16x64x128) per h instead of 100M
// transcendentals.

typedef __attribute__((ext_vector_type(2))) float v2f;
typedef __attribute__((ext_vector_type(8))) float v8f;

#define H_DIM 768
#define N_DIM 64
#define L_DIM 2048
#define T0 128                 // inner block length (l0)
#define T1 16                  // number of outer blocks (l1) = L/T0
#define ASTRIDE 66             // padded row stride for A panels (bank-conflict-free)
#define WSTRIDE 136            // padded row stride for W panels (half-waves hit disjoint banks)

__launch_bounds__(256)
__global__ void s4d_vandermonde_wmma(const float* __restrict__ log_dt,
                                     const float* __restrict__ B_ri,
                                     const float* __restrict__ inv_A_real,
                                     const float* __restrict__ A_imag,
                                     const float* __restrict__ C_ri,
                                     float* __restrict__ Kout)
{
    __shared__ float sAr[T1][ASTRIDE];      // real part of A panel (2x folded)
    __shared__ float sAiN[T1][ASTRIDE];     // NEGATED imag part of A panel
    __shared__ float sWr[N_DIM][WSTRIDE];   // Re(w^l0)
    __shared__ float sWi[N_DIM][WSTRIDE];   // Im(w^l0)
    __shared__ float s_dre[N_DIM], s_dim[N_DIM];  // dtA
    __shared__ float s_ctr[N_DIM], s_cti[N_DIM];  // 2*Ctil (ZOH-discretized C)

    const int h   = blockIdx.x;
    const int tid = threadIdx.x;

    // ---- Phase 1: per-state-n setup (ZOH discretization) ----
    if (tid < N_DIM) {
        const int n = tid;
        const float dt   = __expf(log_dt[h]);
        const float a_re = -__expf(inv_A_real[h * N_DIM + n]);
        const float a_im = A_imag[h * N_DIM + n];
        const float dre  = dt * a_re;
        const float dim  = dt * a_im;

        const float br = B_ri[(h * N_DIM + n) * 2 + 0];
        const float bi = B_ri[(h * N_DIM + n) * 2 + 1];
        const float cr = C_ri[(h * N_DIM + n) * 2 + 0];
        const float ci = C_ri[(h * N_DIM + n) * 2 + 1];
        // Cb = B * C (complex)
        const float cbr = br * cr - bi * ci;
        const float cbi = br * ci + bi * cr;
        // E = exp(dtA) - 1
        const float em = __expf(dre);
        float es, ec;
        __sincosf(dim, &es, &ec);
        const float er = em * ec - 1.0f;
        const float ei = em * es;
        // F = E / A = E * conj(A) / |A|^2   (A never zero: Re(A) < 0)
        const float inv = 1.0f / (a_re * a_re + a_im * a_im);
        const float fr  = (er * a_re + ei * a_im) * inv;
        const float fi  = (ei * a_re - er * a_im) * inv;
        // 2*Ctil = 2 * Cb * F  (fold the 2*Re(...) factor in here)
        s_ctr[n] = 2.0f * (cbr * fr - cbi * fi);
        s_cti[n] = 2.0f * (cbr * fi + cbi * fr);
        s_dre[n] = dre;
        s_dim[n] = dim;
    }
    __syncthreads();

    // ---- Phase 2a: A panel (T1 x N): 2*Ctil_n * w_n^(128*l1), imag negated ----
    for (int idx = tid; idx < T1 * N_DIM; idx += 256) {
        const int l1 = idx >> 6;        // / 64
        const int n  = idx & 63;        // % 64
        const float t = 128.0f * (float)l1;
        const float m = __expf(s_dre[n] * t);
        float s, c;
        __sincosf(s_dim[n] * t, &s, &c);
        const float wr = m * c, wi = m * s;
        const float ctr = s_ctr[n], cti = s_cti[n];
        sAr[l1][n]  =  ctr * wr - cti * wi;
        sAiN[l1][n] = -(ctr * wi + cti * wr);
    }
    // ---- Phase 2b: W panel (N x T0): w_n^(l0) ----
    for (int idx = tid; idx < N_DIM * T0; idx += 256) {
        const int n  = idx >> 7;        // / 128
        const int l0 = idx & 127;       // % 128
        const float t = (float)l0;
        const float m = __expf(s_dre[n] * t);
        float s, c;
        __sincosf(s_dim[n] * t, &s, &c);
        sWr[n][l0] = m * c;
        sWi[n][l0] = m * s;
    }
    __syncthreads();

    // ---- Phase 3: WMMA. Wave j owns 16x16 output tile, cols [16j, 16j+16) ----
    const int wave  = tid >> 5;         // 0..7  (wave32)
    const int lane  = tid & 31;
    const int row   = lane & 15;        // A-row (M) / B-col (N-in-tile)
    const int khalf = lane >> 4;        // 0: K={0,1}; 1: K={2,3}  (per ISA frag layout)
    const int col   = wave * 16 + row;  // l0 column in W panel

    v8f acc = {};
#pragma unroll
    for (int kb = 0; kb < N_DIM / 4; ++kb) {
        const int k0 = 4 * kb + 2 * khalf;   // this lane's first K element
        v2f a, b;
        // real product:  Ar @ Wr
        a.x = sAr[row][k0];
        a.y = sAr[row][k0 + 1];
        b.x = sWr[k0][col];
        b.y = sWr[k0 + 1][col];
        acc = __builtin_amdgcn_wmma_f32_16x16x4_f32(false, a, false, b,
                                                    (short)0, acc, false, false);
        // imag product: (-Ai) @ Wi
        a.x = sAiN[row][k0];
        a.y = sAiN[row][k0 + 1];
        b.x = sWi[k0][col];
        b.y = sWi[k0 + 1][col];
        acc = __builtin_amdgcn_wmma_f32_16x16x4_f32(false, a, false, b,
                                                    (short)0, acc, false, false);
    }

    // ---- Write D: VGPR v holds M = v + 8*(lane>=16), N = lane%16 ----
    float* Kh = Kout + (size_t)h * L_DIM;
#pragma unroll
    for (int v = 0; v < 8; ++v) {
        const int l1 = v + 8 * khalf;
        Kh[l1 * T0 + wave * 16 + row] = acc[v];
    }
}

extern "C" void kernel_launch(void* const* d_in, const int* in_sizes, int n_in,
                              void* d_out, int out_size, void* d_ws, size_t ws_size,
                              hipStream_t stream) {
    (void)in_sizes; (void)n_in; (void)out_size; (void)d_ws; (void)ws_size;
    const float* log_dt     = (const float*)d_in[0];
    const float* B_ri       = (const float*)d_in[1];
    const float* inv_A_real = (const float*)d_in[2];
    const float* A_imag     = (const float*)d_in[3];
    const float* C_ri       = (const float*)d_in[4];
    // d_in[5] is L (==2048), compile-time constant here.
    float* Kout = (float*)d_out;

    dim3 grid(H_DIM);
    dim3 block(256);   // 8 waves of 32 on gfx1250
    s4d_vandermonde_wmma<<<grid, block, 0, stream>>>(log_dt, B_ri, inv_A_real,
                                                     A_imag, C_ri, Kout);
}